// Decimation_25142738551433
// MI455X (gfx1250) — compile-verified
//
#include <hip/hip_runtime.h>
#include <math.h>

#define N_NODES   100000
#define N_EDGES   1600000
#define F_INPUT   128
#define HID       64
#define N_HEADS   8
#define CHEB_K    14
#define N_LAYERS  3
#define N_GRAPHS  128
#define N_CLASSES 10

typedef __attribute__((ext_vector_type(2))) float v2f;
typedef __attribute__((ext_vector_type(8))) float v8f;

// ------------------------- graph preprocessing -------------------------

__global__ void k_zero_i32(int* __restrict__ p, int n) {
  int i = blockIdx.x * blockDim.x + threadIdx.x;
  if (i < n) p[i] = 0;
}

__global__ void k_count_deg(const int* __restrict__ row, int* __restrict__ deg) {
  int i = blockIdx.x * blockDim.x + threadIdx.x;
  if (i < N_EDGES) atomicAdd(&deg[row[i]], 1);
}

__global__ void k_make_dinv(const int* __restrict__ deg, float* __restrict__ dinv) {
  int i = blockIdx.x * blockDim.x + threadIdx.x;
  if (i < N_NODES) {
    float d = (float)deg[i];
    dinv[i] = rsqrtf(d > 1.0f ? d : 1.0f);
  }
}

// Single-workgroup chunked Hillis-Steele exclusive scan -> row_ptr.
// Also zeroes deg[] in place so it can be reused as the fill cursor.
__global__ void k_scan_rowptr(int* __restrict__ deg, int* __restrict__ rp) {
  __shared__ int sdata[1024];
  __shared__ int s_carry;
  int t = threadIdx.x;
  if (t == 0) s_carry = 0;
  __syncthreads();
  for (int base = 0; base < N_NODES; base += 1024) {
    int i = base + t;
    int v = (i < N_NODES) ? deg[i] : 0;
    sdata[t] = v;
    __syncthreads();
    for (int off = 1; off < 1024; off <<= 1) {
      int tmp = (t >= off) ? sdata[t - off] : 0;
      __syncthreads();
      sdata[t] += tmp;
      __syncthreads();
    }
    if (i < N_NODES) {
      rp[i] = s_carry + sdata[t] - v;  // exclusive prefix
      deg[i] = 0;                      // becomes fill cursor
    }
    __syncthreads();
    if (t == 0) s_carry += sdata[1023];
    __syncthreads();
  }
  if (t == 0) rp[N_NODES] = s_carry;   // == N_EDGES
}

__global__ void k_fill_csr(const int* __restrict__ row, const int* __restrict__ col,
                           const float* __restrict__ dinv, const int* __restrict__ rp,
                           int* __restrict__ cursor, int* __restrict__ ecol,
                           float* __restrict__ ew) {
  int i = blockIdx.x * blockDim.x + threadIdx.x;
  if (i < N_EDGES) {
    int r = row[i], c = col[i];
    int slot = rp[r] + atomicAdd(&cursor[r], 1);
    ecol[slot] = c;
    ew[slot]   = dinv[r] * dinv[c];
  }
}

// ------------------------- dense GEMM via f32 WMMA -------------------------
// Y[N x 64] = Hm[N x fin] @ W[fin x 64].  One wave -> one 16x16 output tile,
// V_WMMA_F32_16X16X4_F32 stepping K by 4.  N and HID tile exactly (no tails),
// so EXEC is all-ones for every WMMA as the ISA requires.
__global__ void k_gemm_wmma(const float* __restrict__ Hm, const float* __restrict__ W,
                            float* __restrict__ Y, int fin) {
  int wg   = blockIdx.x * 8 + (threadIdx.x >> 5);  // 8 waves per block
  int lane = threadIdx.x & 31;
  int mt = wg >> 2;                                // HID/16 = 4 n-tiles
  int nt = wg & 3;
  int m0 = mt * 16, n0 = nt * 16;
  int half = lane >> 4, l = lane & 15;
  const float* arow = Hm + (size_t)(m0 + l) * fin + 2 * half;
  v8f c = {};
  for (int k0 = 0; k0 < fin; k0 += 4) {
    // A 16x4 f32: lane l holds (M=l, K=2*half), (M=l, K=2*half+1)
    v2f a = *(const v2f*)(arow + k0);
    // B 4x16 f32: lane l holds (K=2*half, N=l), (K=2*half+1, N=l)
    v2f b;
    b.x = W[(size_t)(k0 + 2 * half + 0) * HID + n0 + l];
    b.y = W[(size_t)(k0 + 2 * half + 1) * HID + n0 + l];
    c = __builtin_amdgcn_wmma_f32_16x16x4_f32(false, a, false, b, (short)0, c,
                                              false, false);
  }
#pragma unroll
  for (int r = 0; r < 8; ++r)   // D: VGPR r -> M = r (lanes 0-15) / 8+r (lanes 16-31)
    Y[(size_t)(m0 + half * 8 + r) * HID + n0 + l] = c[r];
}

// ------------------------- Chebyshev propagation -------------------------
// prop(y)[v] = -sum_{e in row v} w_e * y[col_e]   (CSR gather, no atomics)
// One wave per node, each lane owns 2 of the 64 features (float2).

__device__ __forceinline__ float cheb_coeff(const float* __restrict__ theta, int k) {
  float s = 0.f;
#pragma unroll
  for (int h = 0; h < N_HEADS; ++h) s += theta[h * CHEB_K + k];
  return s * (1.0f / N_HEADS);
}

__global__ void k_cheb_init(const float* __restrict__ Y,
                            const int* __restrict__ rp, const int* __restrict__ ecol,
                            const float* __restrict__ ew, const float* __restrict__ theta,
                            float* __restrict__ Tcur, float* __restrict__ O) {
  int v    = blockIdx.x * 8 + (threadIdx.x >> 5);
  int lane = threadIdx.x & 31;
  int f    = lane * 2;
  float c0 = cheb_coeff(theta, 0);
  float c1 = cheb_coeff(theta, 1);
  int beg = rp[v], end = rp[v + 1];
  float ax = 0.f, ay = 0.f;
  for (int e = beg; e < end; ++e) {
    int   cidx = ecol[e];
    float w    = ew[e];
    float2 yv = *(const float2*)(Y + (size_t)cidx * HID + f);
    ax = fmaf(w, yv.x, ax);
    ay = fmaf(w, yv.y, ay);
  }
  float2 t; t.x = -ax; t.y = -ay;                 // T1 = prop(y)
  *(float2*)(Tcur + (size_t)v * HID + f) = t;
  float2 a = *(const float2*)(Y + (size_t)v * HID + f);
  float2 o;
  o.x = c0 * a.x + c1 * t.x;
  o.y = c0 * a.y + c1 * t.y;
  *(float2*)(O + (size_t)v * HID + f) = o;        // out = c0*T0 + c1*T1
}

__global__ void k_cheb_step(const float* __restrict__ Tcur, const float* __restrict__ Tprev,
                            const int* __restrict__ rp, const int* __restrict__ ecol,
                            const float* __restrict__ ew, const float* __restrict__ theta,
                            int k, float* __restrict__ Tnext, float* __restrict__ O) {
  int v    = blockIdx.x * 8 + (threadIdx.x >> 5);
  int lane = threadIdx.x & 31;
  int f    = lane * 2;
  float ck = cheb_coeff(theta, k);
  int beg = rp[v], end = rp[v + 1];
  float ax = 0.f, ay = 0.f;
  for (int e = beg; e < end; ++e) {
    int   cidx = ecol[e];
    float w    = ew[e];
    float2 yv = *(const float2*)(Tcur + (size_t)cidx * HID + f);
    ax = fmaf(w, yv.x, ax);
    ay = fmaf(w, yv.y, ay);
  }
  float2 tp = *(const float2*)(Tprev + (size_t)v * HID + f);
  float2 tn;                                      // Tk = 2*prop(Tcur) - Tprev
  tn.x = -2.0f * ax - tp.x;
  tn.y = -2.0f * ay - tp.y;
  *(float2*)(Tnext + (size_t)v * HID + f) = tn;
  float2 o = *(float2*)(O + (size_t)v * HID + f);
  o.x = fmaf(ck, tn.x, o.x);
  o.y = fmaf(ck, tn.y, o.y);
  *(float2*)(O + (size_t)v * HID + f) = o;
}

__global__ void k_bias_relu(float* __restrict__ H, const float* __restrict__ b) {
  int i = blockIdx.x * blockDim.x + threadIdx.x;
  if (i < N_NODES * HID) {
    float t = H[i] + b[i & (HID - 1)];
    H[i] = t > 0.f ? t : 0.f;
  }
}

// ------------------------- pooling + MLP head -------------------------

__device__ __forceinline__ int lower_bound_i(const int* __restrict__ a, int n, int val) {
  int lo = 0, hi = n;
  while (lo < hi) {
    int mid = (lo + hi) >> 1;
    if (a[mid] < val) lo = mid + 1; else hi = mid;
  }
  return lo;
}

// One block per graph (batch is sorted): deterministic mean-pool, then the
// 64x64 + 64x10 MLP and log_softmax, fused.
__global__ void k_pool_head(const float* __restrict__ H, const int* __restrict__ batch,
                            const float* __restrict__ l1w, const float* __restrict__ l1b,
                            const float* __restrict__ l2w, const float* __restrict__ l2b,
                            float* __restrict__ out) {
  __shared__ float pooled[HID];
  __shared__ float gv[HID];
  __shared__ float logits[N_CLASSES];
  __shared__ float red[2];
  int g = blockIdx.x;
  int f = threadIdx.x;  // 64 threads
  int lo = lower_bound_i(batch, N_NODES, g);
  int hi = lower_bound_i(batch, N_NODES, g + 1);
  float s = 0.f;
  for (int v = lo; v < hi; ++v) s += H[(size_t)v * HID + f];
  float cnt = (float)(hi - lo);
  if (cnt < 1.f) cnt = 1.f;
  pooled[f] = s / cnt;
  __syncthreads();
  float a = l1b[f];
  for (int j = 0; j < HID; ++j) a = fmaf(pooled[j], l1w[j * HID + f], a);
  gv[f] = a > 0.f ? a : 0.f;
  __syncthreads();
  if (f < N_CLASSES) {
    float b2 = l2b[f];
    for (int j = 0; j < HID; ++j) b2 = fmaf(gv[j], l2w[j * N_CLASSES + f], b2);
    logits[f] = b2;
  }
  __syncthreads();
  if (f == 0) {
    float m = logits[0];
    for (int j = 1; j < N_CLASSES; ++j) m = logits[j] > m ? logits[j] : m;
    float se = 0.f;
    for (int j = 0; j < N_CLASSES; ++j) se += expf(logits[j] - m);
    red[0] = m;
    red[1] = logf(se);
  }
  __syncthreads();
  if (f < N_CLASSES) out[g * N_CLASSES + f] = logits[f] - red[0] - red[1];
}

// ------------------------- host driver -------------------------

extern "C" void kernel_launch(void* const* d_in, const int* in_sizes, int n_in,
                              void* d_out, int out_size, void* d_ws, size_t ws_size,
                              hipStream_t stream) {
  (void)in_sizes; (void)n_in; (void)out_size; (void)ws_size;

  const float* x      = (const float*)d_in[0];
  const int*   eidx   = (const int*)d_in[1];
  const int*   row    = eidx;
  const int*   col    = eidx + N_EDGES;
  const int*   batch  = (const int*)d_in[2];
  const float* W1     = (const float*)d_in[3];
  const float* theta1 = (const float*)d_in[4];
  const float* b1     = (const float*)d_in[5];
  const float* Ws     = (const float*)d_in[6];
  const float* thetas = (const float*)d_in[7];
  const float* bs     = (const float*)d_in[8];
  const float* l1w    = (const float*)d_in[9];
  const float* l1b    = (const float*)d_in[10];
  const float* l2w    = (const float*)d_in[11];
  const float* l2b    = (const float*)d_in[12];
  float*       out    = (float*)d_out;

  char* ws = (char*)d_ws;
  size_t off = 0;
  auto take = [&](size_t bytes) -> void* {
    void* p = ws + off;
    off += (bytes + 255) & ~(size_t)255;
    return p;
  };
  int*   deg  = (int*)  take((size_t)N_NODES * 4);        // deg, then fill cursor
  int*   rp   = (int*)  take((size_t)(N_NODES + 1) * 4);
  float* dinv = (float*)take((size_t)N_NODES * 4);
  int*   ecol = (int*)  take((size_t)N_EDGES * 4);
  float* ew   = (float*)take((size_t)N_EDGES * 4);
  float* bufH = (float*)take((size_t)N_NODES * HID * 4);  // out-acc / layer output
  float* bufA = (float*)take((size_t)N_NODES * HID * 4);
  float* bufB = (float*)take((size_t)N_NODES * HID * 4);
  float* bufC = (float*)take((size_t)N_NODES * HID * 4);

  const int TB = 256;
  // --- build normalized CSR (once) ---
  k_zero_i32  <<<(N_NODES + TB - 1) / TB, TB, 0, stream>>>(deg, N_NODES);
  k_count_deg <<<(N_EDGES + TB - 1) / TB, TB, 0, stream>>>(row, deg);
  k_make_dinv <<<(N_NODES + TB - 1) / TB, TB, 0, stream>>>(deg, dinv);
  k_scan_rowptr<<<1, 1024, 0, stream>>>(deg, rp);
  k_fill_csr  <<<(N_EDGES + TB - 1) / TB, TB, 0, stream>>>(row, col, dinv, rp, deg, ecol, ew);

  // --- spectral layers ---
  const float* hin = x;
  int fin = F_INPUT;
  const int gemm_blocks = (N_NODES / 16) * (HID / 16) / 8;  // 3125
  const int node_blocks = N_NODES / 8;                      // 12500 (wave per node)
  for (int layer = 0; layer < N_LAYERS; ++layer) {
    const float* Wp = (layer == 0) ? W1     : Ws     + (size_t)(layer - 1) * HID * HID;
    const float* th = (layer == 0) ? theta1 : thetas + (size_t)(layer - 1) * N_HEADS * CHEB_K;
    const float* bp = (layer == 0) ? b1     : bs     + (size_t)(layer - 1) * HID;

    k_gemm_wmma<<<gemm_blocks, TB, 0, stream>>>(hin, Wp, bufA, fin);      // T0 = h@W
    k_cheb_init<<<node_blocks, TB, 0, stream>>>(bufA, rp, ecol, ew, th, bufB, bufH);
    float* tp = bufA; float* tc = bufB; float* tn = bufC;
    for (int k = 2; k < CHEB_K; ++k) {
      k_cheb_step<<<node_blocks, TB, 0, stream>>>(tc, tp, rp, ecol, ew, th, k, tn, bufH);
      float* tmp = tp; tp = tc; tc = tn; tn = tmp;
    }
    k_bias_relu<<<(N_NODES * HID + TB - 1) / TB, TB, 0, stream>>>(bufH, bp);
    hin = bufH;
    fin = HID;
  }

  // --- pool + head ---
  k_pool_head<<<N_GRAPHS, HID, 0, stream>>>(bufH, batch, l1w, l1b, l2w, l2b, out);
}